// SGConvClassification_80418967650354
// MI455X (gfx1250) — compile-verified
//
#include <hip/hip_runtime.h>
#include <math.h>

#define N_NODES    100000
#define N_EDGES    3200000
#define N_FEAT     128
#define N_CLASSES  16
#define NUM_GRAPHS 512

typedef __attribute__((ext_vector_type(2))) float v2f;
typedef __attribute__((ext_vector_type(8))) float v8f;

__global__ void fill_f32(float* __restrict__ p, float v, int n) {
  int i = blockIdx.x * blockDim.x + threadIdx.x;
  if (i < n) p[i] = v;
}

// deg[dst] += 1 for every real edge (deg pre-filled with 1.0 for self-loop)
__global__ void deg_accum(const int* __restrict__ dst, float* __restrict__ deg, int E) {
  int e = blockIdx.x * blockDim.x + threadIdx.x;
  if (e < E) atomicAdd(&deg[dst[e]], 1.0f);
}

__global__ void to_dinv(float* __restrict__ deg, int n) {
  int i = blockIdx.x * blockDim.x + threadIdx.x;
  if (i < n) {
    float d = deg[i];
    deg[i] = (d > 0.0f) ? (1.0f / sqrtf(d)) : 0.0f;
  }
}

// y[N,16] = x[N,128] @ W[128,16], one 16-row tile per wave, f32 WMMA, K split in 4s.
__global__ void gemm_xw_wmma(const float* __restrict__ x, const float* __restrict__ W,
                             float* __restrict__ y, int numTiles) {
  int wave = (blockIdx.x * blockDim.x + threadIdx.x) >> 5;
  if (wave >= numTiles) return;               // wave-uniform: EXEC stays all-1s
  int lane = threadIdx.x & 31;
  int nc   = lane & 15;                       // A row (M) / B,D col (N)
  int koff = (lane >> 4) << 1;                // lanes 0-15 -> K{0,1}, 16-31 -> K{2,3}
  const float* xr = x + (size_t)(wave * 16 + nc) * N_FEAT;
  v8f c = {};
  #pragma unroll 8
  for (int kk = 0; kk < N_FEAT; kk += 4) {
    v2f a = *(const v2f*)(xr + kk + koff);    // contiguous K pair, 8B load
    v2f bm;
    bm.x = W[(kk + koff    ) * N_CLASSES + nc];
    bm.y = W[(kk + koff + 1) * N_CLASSES + nc];
    c = __builtin_amdgcn_wmma_f32_16x16x4_f32(false, a, false, bm,
                                              (short)0, c, false, false);
  }
  // D layout: VGPR j -> M=j (lanes 0-15) / M=j+8 (lanes 16-31), N=lane&15
  int mbase = wave * 16 + ((lane < 16) ? 0 : 8);
  #pragma unroll
  for (int j = 0; j < 8; ++j)
    y[(size_t)(mbase + j) * N_CLASSES + nc] = c[j];
}

// out[dst] += dinv[src]*dinv[dst] * in[src], 16 features: 16 lanes per edge.
__global__ void propagate(const float* __restrict__ in, float* __restrict__ out,
                          const int* __restrict__ src, const int* __restrict__ dst,
                          const float* __restrict__ dinv, int E, int N) {
  int idx = blockIdx.x * blockDim.x + threadIdx.x;
  int e = idx >> 4;
  int f = idx & 15;
  if (e >= E + N) return;
  int s, d;
  if (e < E) { s = src[e]; d = dst[e]; }
  else       { s = d = e - E; }               // self-loop
  float nrm = dinv[s] * dinv[d];
  atomicAdd(&out[(size_t)d * N_CLASSES + f], nrm * in[(size_t)s * N_CLASSES + f]);
}

__global__ void pool(const float* __restrict__ in, const int* __restrict__ batch,
                     float* __restrict__ sums, float* __restrict__ cnts, int N) {
  int idx = blockIdx.x * blockDim.x + threadIdx.x;
  int n = idx >> 4, f = idx & 15;
  if (n >= N) return;
  int g = batch[n];
  atomicAdd(&sums[g * N_CLASSES + f], in[(size_t)n * N_CLASSES + f]);
  if (f == 0) atomicAdd(&cnts[g], 1.0f);
}

__global__ void logits_lsm(const float* __restrict__ sums, const float* __restrict__ cnts,
                           const float* __restrict__ b, float* __restrict__ out, int G) {
  int g = blockIdx.x * blockDim.x + threadIdx.x;
  if (g >= G) return;
  float inv = 1.0f / fmaxf(cnts[g], 1.0f);
  float z[N_CLASSES];
  float mx = -INFINITY;
  #pragma unroll
  for (int c = 0; c < N_CLASSES; ++c) {
    z[c] = sums[g * N_CLASSES + c] * inv + b[c];   // bias folded in post-pool (linear)
    mx = fmaxf(mx, z[c]);
  }
  float se = 0.0f;
  #pragma unroll
  for (int c = 0; c < N_CLASSES; ++c) se += expf(z[c] - mx);
  float lse = logf(se) + mx;
  #pragma unroll
  for (int c = 0; c < N_CLASSES; ++c) out[g * N_CLASSES + c] = z[c] - lse;
}

extern "C" void kernel_launch(void* const* d_in, const int* in_sizes, int n_in,
                              void* d_out, int out_size, void* d_ws, size_t ws_size,
                              hipStream_t stream) {
  const float* x     = (const float*)d_in[0];
  const float* W     = (const float*)d_in[1];
  const float* bias  = (const float*)d_in[2];
  const int*   ei    = (const int*)d_in[3];   // [2, E]
  const int*   batch = (const int*)d_in[4];
  const int*   src   = ei;
  const int*   dst   = ei + N_EDGES;
  float*       out   = (float*)d_out;

  // Workspace layout (floats): dinv[N] | bufA[N*16] | bufB[N*16] | sums[512*16] | cnts[512]
  float* dinv = (float*)d_ws;
  float* bufA = dinv + N_NODES;
  float* bufB = bufA + (size_t)N_NODES * N_CLASSES;
  float* sums = bufB + (size_t)N_NODES * N_CLASSES;
  float* cnts = sums + NUM_GRAPHS * N_CLASSES;

  const int T = 256;
  int nodeBlocks  = (N_NODES + T - 1) / T;
  int featN       = N_NODES * N_CLASSES;
  int featBlocks  = (featN + T - 1) / T;
  int propThreads = (N_EDGES + N_NODES) * 16;
  int propBlocks  = (propThreads + T - 1) / T;

  // 1) degrees (self-loop contributes 1) -> dinv in place
  fill_f32<<<nodeBlocks, T, 0, stream>>>(dinv, 1.0f, N_NODES);
  deg_accum<<<(N_EDGES + T - 1) / T, T, 0, stream>>>(dst, dinv, N_EDGES);
  to_dinv<<<nodeBlocks, T, 0, stream>>>(dinv, N_NODES);

  // 2) GEMM first (linear ops commute): bufA = x @ W  -> features shrink 128 -> 16
  int numTiles = N_NODES / 16;                // 100000 / 16 = 6250 exactly
  int wavesPerBlock = T / 32;
  gemm_xw_wmma<<<(numTiles + wavesPerBlock - 1) / wavesPerBlock, T, 0, stream>>>(
      x, W, bufA, numTiles);

  // 3) three hops of S, ping-pong A<->B
  fill_f32<<<featBlocks, T, 0, stream>>>(bufB, 0.0f, featN);
  propagate<<<propBlocks, T, 0, stream>>>(bufA, bufB, src, dst, dinv, N_EDGES, N_NODES);
  fill_f32<<<featBlocks, T, 0, stream>>>(bufA, 0.0f, featN);
  propagate<<<propBlocks, T, 0, stream>>>(bufB, bufA, src, dst, dinv, N_EDGES, N_NODES);
  fill_f32<<<featBlocks, T, 0, stream>>>(bufB, 0.0f, featN);
  propagate<<<propBlocks, T, 0, stream>>>(bufA, bufB, src, dst, dinv, N_EDGES, N_NODES);

  // 4) mean pool + bias + log_softmax
  fill_f32<<<(NUM_GRAPHS * (N_CLASSES + 1) + T - 1) / T, T, 0, stream>>>(
      sums, 0.0f, NUM_GRAPHS * (N_CLASSES + 1));   // sums + cnts contiguous
  pool<<<featBlocks, T, 0, stream>>>(bufB, batch, sums, cnts, N_NODES);
  logits_lsm<<<(NUM_GRAPHS + T - 1) / T, T, 0, stream>>>(sums, cnts, bias, out, NUM_GRAPHS);
}